// VanillaSoftcapAttention_84567906058329
// MI455X (gfx1250) — compile-verified
//
#include <hip/hip_runtime.h>
#include <hip/hip_bf16.h>

typedef __attribute__((ext_vector_type(2))) float v2f;
typedef __attribute__((ext_vector_type(4))) float v4f;
typedef __attribute__((ext_vector_type(8))) float v8f;

#define L_SEQ   2048
#define HD      64
#define NH      16
#define NB      2
#define DMODEL  (NH * HD)
#define QSCALE  0.125f
#define CAP     50.0f

// ---- wave32 16-lane-group reductions (rows live across lanes 0-15 / 16-31) ----
__device__ __forceinline__ float red_max16(float v) {
    v = fmaxf(v, __shfl_xor(v, 1));
    v = fmaxf(v, __shfl_xor(v, 2));
    v = fmaxf(v, __shfl_xor(v, 4));
    v = fmaxf(v, __shfl_xor(v, 8));
    return v;
}
__device__ __forceinline__ float red_sum16(float v) {
    v += __shfl_xor(v, 1);
    v += __shfl_xor(v, 2);
    v += __shfl_xor(v, 4);
    v += __shfl_xor(v, 8);
    return v;
}

// soft cap: CAP * tanh(s/CAP) = CAP * (1 - 2/(exp(2s/CAP)+1)) ; uses v_exp_f32
__device__ __forceinline__ float softcap(float s) {
    float e = __expf(s * (2.0f / CAP));
    return CAP * (1.0f - 2.0f / (e + 1.0f));
}

// One 16x16 fp32 score tile: S = Qtile(16x64) * K(j0..j0+15, :)^T
// via 16x V_WMMA_F32_16X16X4_F32 along the D=64 contraction.
__device__ __forceinline__ v8f qk_tile(const float* __restrict__ kb,
                                       const v2f* qf, int j0, int lane) {
    const int n  = lane & 15;          // B-matrix column = key within tile
    const int h2 = (lane >> 4) << 1;   // k-pair select per half-wave
    const float* krow = kb + (size_t)(j0 + n) * HD;
    v2f kf[16];
#pragma unroll
    for (int t = 0; t < 16; ++t)
        kf[t] = *(const v2f*)(krow + 4 * t + h2);   // 8B aligned
    v8f s = {0.f, 0.f, 0.f, 0.f, 0.f, 0.f, 0.f, 0.f};
#pragma unroll
    for (int t = 0; t < 16; ++t)
        s = __builtin_amdgcn_wmma_f32_16x16x4_f32(false, qf[t], false, kf[t],
                                                  (short)0, s, false, false);
    return s;
}

__global__ __launch_bounds__(128)
void softcap_attn_kernel(const float* __restrict__ q,
                         const float* __restrict__ k,
                         const float* __restrict__ v,
                         float* __restrict__ out,
                         float* __restrict__ probs) {
    __shared__ float ptile[4 * 16 * 18];   // per-wave 16x16 P tile, stride 18 (even, conflict-free)

    const int lane = threadIdx.x & 31;
    const int wid  = threadIdx.x >> 5;
    const int qblk = blockIdx.x % (L_SEQ / 64);
    const int bh   = blockIdx.x / (L_SEQ / 64);
    const int b    = bh / NH;
    const int h    = bh % NH;
    const int q0   = qblk * 64 + wid * 16;      // this wave's 16 query rows

    const float* qb = q + (size_t)bh * L_SEQ * HD;
    const float* kb = k + (size_t)bh * L_SEQ * HD;
    const float* vb = v + (size_t)bh * L_SEQ * HD;
    float*       pb = probs + (size_t)bh * L_SEQ * L_SEQ;

    const int n  = lane & 15;       // column-in-tile / A-matrix row
    const int hi = lane >> 4;       // half-wave select
    const int h2 = hi << 1;

    // ---- load Q tile as 16 A-fragments (16x4 fp32), pre-scaled ----
    v2f qf[16];
    {
        const float* qrow = qb + (size_t)(q0 + n) * HD;
#pragma unroll
        for (int t = 0; t < 16; ++t) {
            v2f tmp = *(const v2f*)(qrow + 4 * t + h2);
            qf[t] = tmp * QSCALE;
        }
    }

    // Per-lane row stats; VGPR r of the C-layout tile holds row (r + 8*hi).
    float m8[8], l8[8];
#pragma unroll
    for (int r = 0; r < 8; ++r) { m8[r] = -3.0e38f; l8[r] = 0.f; }

    const int ntiles = q0 / 16 + 1;     // causal: key tiles j0 <= q0

    // ================= Pass 1: online row max + sum(exp) =================
    for (int jt = 0; jt < ntiles; ++jt) {
        const int j0 = jt * 16;
        v8f s = qk_tile(kb, qf, j0, lane);
        const bool diag = (j0 == q0);
#pragma unroll
        for (int r = 0; r < 8; ++r) {
            float x = softcap(s[r]);
            if (diag && (n > r + 8 * hi)) x = -3.0e38f;   // causal mask inside diagonal tile
            float mn = fmaxf(m8[r], red_max16(x));
            float e  = __expf(x - mn);
            float es = red_sum16(e);
            l8[r] = l8[r] * __expf(m8[r] - mn) + es;
            m8[r] = mn;
        }
    }

    float rl8[8];
#pragma unroll
    for (int r = 0; r < 8; ++r) rl8[r] = 1.0f / l8[r];

    // ================= Pass 2: probs write + O = P*V =================
    const v8f vzero = {0.f, 0.f, 0.f, 0.f, 0.f, 0.f, 0.f, 0.f};
    v8f o[4];
#pragma unroll
    for (int nt = 0; nt < 4; ++nt) o[nt] = vzero;

    float* myLds = ptile + wid * (16 * 18);

    for (int jt = 0; jt < ntiles; ++jt) {
        const int j0 = jt * 16;
        v8f s = qk_tile(kb, qf, j0, lane);
        const bool diag = (j0 == q0);
#pragma unroll
        for (int r = 0; r < 8; ++r) {
            float x = softcap(s[r]);
            if (diag && (n > r + 8 * hi)) x = -3.0e38f;
            float p = __expf(x - m8[r]) * rl8[r];         // masked -> 0
            // streaming write of probs (non-temporal: keep K/V resident in L2)
            __builtin_nontemporal_store(
                p, pb + (size_t)(q0 + r + 8 * hi) * L_SEQ + j0 + n);
            myLds[(r + 8 * hi) * 18 + n] = p;             // stage for A-layout refragment
        }
        // same-wave LDS store->load ordering; also a compiler reorder fence
        asm volatile("s_wait_dscnt 0" ::: "memory");

        // re-fragment P into A layout (16x4 slices along key dim)
        v2f pf[4];
#pragma unroll
        for (int kk = 0; kk < 4; ++kk)
            pf[kk] = *(const v2f*)(myLds + n * 18 + 4 * kk + h2);   // ds_load_b64

        // O(16x64) += P(16x16) * V(16x64), 4 n-tiles x 4 k-steps
        const float* vt = vb + (size_t)j0 * HD;
#pragma unroll
        for (int nt = 0; nt < 4; ++nt) {
#pragma unroll
            for (int kk = 0; kk < 4; ++kk) {
                const int kidx = 4 * kk + h2;
                v2f bf;
                bf.x = vt[(size_t)kidx * HD + nt * 16 + n];
                bf.y = vt[(size_t)(kidx + 1) * HD + nt * 16 + n];
                o[nt] = __builtin_amdgcn_wmma_f32_16x16x4_f32(
                    false, pf[kk], false, bf, (short)0, o[nt], false, false);
            }
        }
    }

    // ---- zero-fill the strictly-upper (masked) probs tiles with b128 NT stores ----
    {
        const v4f z = {0.f, 0.f, 0.f, 0.f};
        const int zrow = lane >> 2;            // 0..7
        const int zcol = (lane & 3) << 2;      // 0,4,8,12
        for (int j0 = q0 + 16; j0 < L_SEQ; j0 += 16) {
            float* p0 = pb + (size_t)(q0 + zrow) * L_SEQ + j0 + zcol;
            __builtin_nontemporal_store(z, (v4f*)p0);
            __builtin_nontemporal_store(z, (v4f*)(p0 + (size_t)8 * L_SEQ));
        }
    }

    // ---- write O: out[b, q, h*64 + d] ----
    float* ob = out + ((size_t)b * L_SEQ + q0) * DMODEL + h * HD;
#pragma unroll
    for (int nt = 0; nt < 4; ++nt) {
#pragma unroll
        for (int r = 0; r < 8; ++r) {
            ob[(size_t)(r + 8 * hi) * DMODEL + nt * 16 + n] = o[nt][r];
        }
    }
}

extern "C" void kernel_launch(void* const* d_in, const int* in_sizes, int n_in,
                              void* d_out, int out_size, void* d_ws, size_t ws_size,
                              hipStream_t stream) {
    const float* q = (const float*)d_in[0];
    const float* k = (const float*)d_in[1];
    const float* v = (const float*)d_in[2];
    // d_in[3] is the boolean causal mask; causality is baked into the kernel.
    float* out   = (float*)d_out;
    float* probs = out + (size_t)NB * L_SEQ * DMODEL;   // tuple output: [out | probs]

    dim3 grid(NB * NH * (L_SEQ / 64));
    dim3 block(128);
    softcap_attn_kernel<<<grid, block, 0, stream>>>(q, k, v, out, probs);
}